// WindowAttention_54795192762630
// MI455X (gfx1250) — compile-verified
//
#include <hip/hip_runtime.h>
#include <math.h>

typedef __attribute__((ext_vector_type(2))) float v2f;
typedef __attribute__((ext_vector_type(8))) float v8f;
typedef __attribute__((ext_vector_type(4))) unsigned int v4u;
typedef __attribute__((ext_vector_type(4))) int v4i;
typedef __attribute__((ext_vector_type(8))) int v8i;

#define WMMA_F32X4(a, b, c) \
  __builtin_amdgcn_wmma_f32_16x16x4_f32(false, (a), false, (b), (short)0, (c), false, false)

// Problem constants
#define NB    512      // windows (B_)
#define NW    144      // tokens per window (W)
#define NC    512      // channels (C)
#define NH    16       // heads
#define HD    32       // head dim
#define NTOK  (NB*NW)  // 73728 tokens

// Tensor Data Mover availability (5-arg on ROCm7.2/clang-22, 6-arg on clang-23+)
#if defined(__has_builtin)
#  if __has_builtin(__builtin_amdgcn_tensor_load_to_lds)
#    define USE_TDM 1
#  endif
#endif
#ifndef USE_TDM
#  define USE_TDM 0
#endif

// ---------------------------------------------------------------------------
// Kernel 1: qkv = x @ qkv_w^T + qkv_b, scattered into q/k/v (B,H,W,hd) layout.
// Each wave owns a 32x32 output block (2x2 WMMA tiles): 4 loads feed 4 WMMAs,
// doubling FLOP/byte vs a single-tile wave (qkv GEMM carries 116 of 177 GFLOP).
// f32 frag layout (ISA 7.12.2): A 16x4: lane&15 -> M, VGPR v -> K = v + (lane>>4)*2.
//                               B 4x16:  lane&15 -> N, same K mapping. B[k][n] = W[n][k].
// ---------------------------------------------------------------------------
__global__ __launch_bounds__(256) void qkv_gemm_wmma(
    const float* __restrict__ X, const float* __restrict__ Wqkv,
    const float* __restrict__ Bqkv,
    float* __restrict__ q, float* __restrict__ k, float* __restrict__ v)
{
  const int lane = threadIdx.x & 31;
  const int wave = threadIdx.x >> 5;
  const int nt   = (blockIdx.y * 8 + wave) * 2;   // first of 2 N-tiles (0..94)
  const int m    = lane & 15;
  const int half = lane >> 4;
  const int krel = half * 2;

  const float* a0 = X    + (size_t)(blockIdx.x * 32 + m) * NC;
  const float* a1 = a0 + (size_t)16 * NC;
  const float* b0 = Wqkv + (size_t)(nt * 16 + m) * NC;
  const float* b1 = b0 + (size_t)16 * NC;

  v8f acc[2][2] = {};
  for (int k0 = 0; k0 < NC; k0 += 4) {
    v2f A0 = *(const v2f*)(a0 + k0 + krel);
    v2f A1 = *(const v2f*)(a1 + k0 + krel);
    v2f B0 = *(const v2f*)(b0 + k0 + krel);
    v2f B1 = *(const v2f*)(b1 + k0 + krel);
    acc[0][0] = WMMA_F32X4(A0, B0, acc[0][0]);
    acc[0][1] = WMMA_F32X4(A0, B1, acc[0][1]);
    acc[1][0] = WMMA_F32X4(A1, B0, acc[1][0]);
    acc[1][1] = WMMA_F32X4(A1, B1, acc[1][1]);
  }

#pragma unroll
  for (int ni = 0; ni < 2; ++ni) {
    const int col = (nt + ni) * 16 + m;          // 0..1535
    const int sec = col >> 9;                    // 0=q 1=k 2=v
    const int cc  = col & 511;
    const int h   = cc >> 5;
    const int d   = cc & 31;
    const float bias = Bqkv[col];
    float* dst = (sec == 0) ? q : ((sec == 1) ? k : v);
#pragma unroll
    for (int mi = 0; mi < 2; ++mi) {
#pragma unroll
      for (int r = 0; r < 8; ++r) {
        int tok = blockIdx.x * 32 + mi * 16 + r + half * 8;
        int b_  = tok / NW;
        int w_  = tok - b_ * NW;
        dst[(((size_t)b_ * NH + h) * NW + w_) * HD + d] = acc[mi][ni][r] + bias;
      }
    }
  }
}

// ---------------------------------------------------------------------------
// Kernel 2: L2-normalize rows of q and k along hd=32 (one wave32 per row).
// ---------------------------------------------------------------------------
__global__ __launch_bounds__(256) void norm_rows(float* __restrict__ q,
                                                 float* __restrict__ k,
                                                 int rowsPerTensor)
{
  const int lane = threadIdx.x & 31;
  const int wave = threadIdx.x >> 5;
  long long row = (long long)blockIdx.x * 8 + wave;    // 0 .. 2*rows-1
  float* base = (row < rowsPerTensor) ? (q + row * HD)
                                      : (k + (row - rowsPerTensor) * HD);
  float x = base[lane];
  float s = x * x;
  for (int m = 16; m; m >>= 1) s += __shfl_xor(s, m, 32);
  float inv = 1.0f / fmaxf(sqrtf(s), 1e-12f);
  base[lane] = x * inv;
}

// ---------------------------------------------------------------------------
// Kernel 3: per (b,h): S = qn@kn^T * scale + earth_bias + mask; softmax; O = S@v.
// q/k/v staged to LDS by the Tensor Data Mover (pad feature produces the
// bank-conflict-free stride-34 layout directly); S lives in LDS (stride 146,
// even => all v2f frag loads 8B-aligned).
// ---------------------------------------------------------------------------
#define QS_STRIDE 34
#define S_STRIDE  146
#define LDS_S_FLOATS   (NW * S_STRIDE)        // 21024
#define LDS_QKV_FLOATS (NW * QS_STRIDE)       // 4896
#define LDS_TOTAL_BYTES ((LDS_S_FLOATS + 3 * LDS_QKV_FLOATS) * 4)  // 142848
#define LDS_Q_BYTEOFF  (LDS_S_FLOATS * 4)                    // 84096
#define LDS_K_BYTEOFF  (LDS_Q_BYTEOFF + LDS_QKV_FLOATS * 4)  // 103680
#define LDS_V_BYTEOFF  (LDS_K_BYTEOFF + LDS_QKV_FLOATS * 4)  // 123264

#if USE_TDM
// D# per ISA 8.3/8.4: 2-D tensor, 144x32 fp32 tile, contiguous rows (stride 32
// elems). pad_enable with interval=32 dwords (code 4) + amount=2 dwords (code 1)
// => LDS rows land at stride 34 floats. Groups 2/3 zero (<=2D). type=2 (image).
__device__ __forceinline__ void tdm_load_qkv_tile(const float* gptr,
                                                  unsigned lds_byte_off)
{
  unsigned long long ga = (unsigned long long)(uintptr_t)gptr;
  v4u g0;
  g0.x = 1u;                                    // count=1 valid descriptor
  g0.y = lds_byte_off;                          // lds_addr (bytes)
  g0.z = (unsigned)(ga & 0xffffffffu);          // global_addr[31:0]
  g0.w = (unsigned)((ga >> 32) & 0x01ffffffu)   // global_addr[56:32]
         | (2u << 30);                          // type=2
  v8i g1;
  g1[0] = (int)0x03120000u;  // mask=0|data_size=2(4B)|pad_en|pad_int=4(32dw)|pad_amt=1(2dw)
  g1[1] = (int)(32u  << 16); // tensor_dim0 = 32 elems (bits 63:48)
  g1[2] = (int)(144u << 16); // tensor_dim1 = 144      (bits 95:80)
  g1[3] = (int)(32u  << 16); // tile_dim0   = 32       (bits 127:112)
  g1[4] = 144;               // tile_dim1   = 144      (bits 143:128)
  g1[5] = 32;                // tensor_dim0_stride = 32 elems (bits 191:160)
  g1[6] = 0;
  g1[7] = 0;
  v4i z = {0, 0, 0, 0};
#if __clang_major__ >= 23
  v8i z8 = {0, 0, 0, 0, 0, 0, 0, 0};
  __builtin_amdgcn_tensor_load_to_lds(g0, g1, z, z, z8, 0);
#else
  __builtin_amdgcn_tensor_load_to_lds(g0, g1, z, z, 0);
#endif
}
#endif

__global__ __launch_bounds__(256) void attn_wmma(
    const float* __restrict__ qn, const float* __restrict__ kn,
    const float* __restrict__ vv, const float* __restrict__ logit_scale,
    const float* __restrict__ earth_bias, const float* __restrict__ mask,
    float* __restrict__ y2)
{
  extern __shared__ float smem[];
  float* Sm = smem;                               // 144 x 146
  float* Qs = smem + LDS_S_FLOATS;                // 144 x 34
  float* Ks = Qs + LDS_QKV_FLOATS;
  float* Vs = Ks + LDS_QKV_FLOATS;

  const int bh   = blockIdx.x;      // 0..8191
  const int b    = bh >> 4;
  const int h    = bh & 15;
  const int tid  = threadIdx.x;
  const int lane = tid & 31;
  const int wave = tid >> 5;
  const int m    = lane & 15;
  const int half = lane >> 4;
  const int krel = half * 2;

  const size_t base = ((size_t)b * NH + h) * NW * HD;

#if USE_TDM
  if (wave == 0) {                       // TDM ignores EXEC; one wave, 3 DMAs
    tdm_load_qkv_tile(qn + base, LDS_Q_BYTEOFF);
    tdm_load_qkv_tile(kn + base, LDS_K_BYTEOFF);
    tdm_load_qkv_tile(vv + base, LDS_V_BYTEOFF);
    __builtin_amdgcn_s_wait_tensorcnt(0);
  }
  __syncthreads();
#else
  for (int e = tid; e < NW * HD; e += 256) {      // 4608 = 18*256
    int r = e >> 5, d = e & 31;
    Qs[r * QS_STRIDE + d] = qn[base + e];
    Ks[r * QS_STRIDE + d] = kn[base + e];
    Vs[r * QS_STRIDE + d] = vv[base + e];
  }
  __syncthreads();
#endif

  const float scale = __expf(fminf(logit_scale[h], 4.6051702f)); // log(1/0.01)
  const float* eb = earth_bias + (size_t)b * NW * NW;
  const float* mk = mask       + (size_t)b * NW * NW;

  // --- S = qn @ kn^T (9x9 tiles of 16x16, K=32) ---
  for (int t = wave; t < 81; t += 8) {
    int ti = t / 9, tj = t - ti * 9;
    const float* qrow = Qs + (ti * 16 + m) * QS_STRIDE;
    const float* krow = Ks + (tj * 16 + m) * QS_STRIDE;
    v8f c = {};
    for (int k0 = 0; k0 < HD; k0 += 4) {
      v2f a  = *(const v2f*)(qrow + k0 + krel);
      v2f bf = *(const v2f*)(krow + k0 + krel);
      c = WMMA_F32X4(a, bf, c);
    }
    int coln = tj * 16 + m;
    for (int r = 0; r < 8; ++r) {
      int rown = ti * 16 + r + half * 8;
      Sm[rown * S_STRIDE + coln] =
          c[r] * scale + eb[rown * NW + coln] + mk[rown * NW + coln];
    }
  }
  __syncthreads();

  // --- softmax along rows (one wave32 per row) ---
  for (int row = wave; row < NW; row += 8) {
    float* Sr = Sm + row * S_STRIDE;
    float mx = -1e30f;
    for (int j = lane; j < NW; j += 32) mx = fmaxf(mx, Sr[j]);
    for (int s = 16; s; s >>= 1) mx = fmaxf(mx, __shfl_xor(mx, s, 32));
    float sum = 0.0f;
    for (int j = lane; j < NW; j += 32) { float e = __expf(Sr[j] - mx); Sr[j] = e; sum += e; }
    for (int s = 16; s; s >>= 1) sum += __shfl_xor(sum, s, 32);
    float inv = 1.0f / sum;
    for (int j = lane; j < NW; j += 32) Sr[j] *= inv;
  }
  __syncthreads();

  // --- O = P @ V (9x2 tiles, K=144) -> y2[b][w][h*32+d] ---
  for (int t = wave; t < 18; t += 8) {
    int ti = t >> 1, tj = t & 1;
    const float* prow = Sm + (ti * 16 + m) * S_STRIDE;
    const int dcol = tj * 16 + m;
    v8f c = {};
    for (int k0 = 0; k0 < NW; k0 += 4) {
      int kk = k0 + krel;
      v2f a = *(const v2f*)(prow + kk);
      v2f bf;
      bf.x = Vs[kk * QS_STRIDE + dcol];
      bf.y = Vs[(kk + 1) * QS_STRIDE + dcol];
      c = WMMA_F32X4(a, bf, c);
    }
    for (int r = 0; r < 8; ++r) {
      int wrow = ti * 16 + r + half * 8;
      y2[((size_t)b * NW + wrow) * NC + h * HD + dcol] = c[r];
    }
  }
}

// ---------------------------------------------------------------------------
// Kernel 4: out = y2 @ proj_w^T + proj_b (2x2-tile WMMA GEMM like kernel 1)
// ---------------------------------------------------------------------------
__global__ __launch_bounds__(256) void proj_gemm_wmma(
    const float* __restrict__ Y, const float* __restrict__ Wp,
    const float* __restrict__ Bp, float* __restrict__ out)
{
  const int lane = threadIdx.x & 31;
  const int wave = threadIdx.x >> 5;
  const int nt   = (blockIdx.y * 8 + wave) * 2;   // 0..30
  const int m    = lane & 15;
  const int half = lane >> 4;
  const int krel = half * 2;

  const float* a0 = Y  + (size_t)(blockIdx.x * 32 + m) * NC;
  const float* a1 = a0 + (size_t)16 * NC;
  const float* b0 = Wp + (size_t)(nt * 16 + m) * NC;
  const float* b1 = b0 + (size_t)16 * NC;

  v8f acc[2][2] = {};
  for (int k0 = 0; k0 < NC; k0 += 4) {
    v2f A0 = *(const v2f*)(a0 + k0 + krel);
    v2f A1 = *(const v2f*)(a1 + k0 + krel);
    v2f B0 = *(const v2f*)(b0 + k0 + krel);
    v2f B1 = *(const v2f*)(b1 + k0 + krel);
    acc[0][0] = WMMA_F32X4(A0, B0, acc[0][0]);
    acc[0][1] = WMMA_F32X4(A0, B1, acc[0][1]);
    acc[1][0] = WMMA_F32X4(A1, B0, acc[1][0]);
    acc[1][1] = WMMA_F32X4(A1, B1, acc[1][1]);
  }

#pragma unroll
  for (int ni = 0; ni < 2; ++ni) {
    const int col = (nt + ni) * 16 + m;
    const float bias = Bp[col];
#pragma unroll
    for (int mi = 0; mi < 2; ++mi) {
#pragma unroll
      for (int r = 0; r < 8; ++r) {
        size_t row = (size_t)blockIdx.x * 32 + mi * 16 + r + half * 8;
        out[row * NC + col] = acc[mi][ni][r] + bias;
      }
    }
  }
}

// ---------------------------------------------------------------------------
extern "C" void kernel_launch(void* const* d_in, const int* in_sizes, int n_in,
                              void* d_out, int out_size, void* d_ws, size_t ws_size,
                              hipStream_t stream)
{
  (void)in_sizes; (void)n_in; (void)out_size; (void)ws_size;
  const float* x           = (const float*)d_in[0];
  const float* mask        = (const float*)d_in[1];
  const float* qkv_w       = (const float*)d_in[2];
  const float* qkv_b       = (const float*)d_in[3];
  const float* proj_w      = (const float*)d_in[4];
  const float* proj_b      = (const float*)d_in[5];
  const float* logit_scale = (const float*)d_in[6];
  const float* earth_bias  = (const float*)d_in[7];
  float* out = (float*)d_out;

  const size_t QKV_ELEMS = (size_t)NB * NH * NW * HD;   // 37,748,736 floats each
  float* q  = (float*)d_ws;
  float* k  = q  + QKV_ELEMS;
  float* v  = k  + QKV_ELEMS;
  float* y2 = v  + QKV_ELEMS;                           // (B_, W, C) attn output

  // 1) qkv projection: waves own 32x32 blocks; N-groups = 1536/32 = 48 = 8*6
  qkv_gemm_wmma<<<dim3(NTOK / 32, 6), 256, 0, stream>>>(x, qkv_w, qkv_b, q, k, v);

  // 2) cosine normalization of q, k
  const int rowsPerTensor = (int)(QKV_ELEMS / HD);      // 1,179,648
  norm_rows<<<dim3((2 * rowsPerTensor) / 8), 256, 0, stream>>>(q, k, rowsPerTensor);

  // 3) attention per (window, head): TDM-staged LDS, fused scale/bias/mask+softmax
  attn_wmma<<<dim3(NB * NH), 256, LDS_TOTAL_BYTES, stream>>>(
      q, k, v, logit_scale, earth_bias, mask, y2);

  // 4) output projection: N-groups = 512/32 = 16 = 8*2
  proj_gemm_wmma<<<dim3(NTOK / 32, 2), 256, 0, stream>>>(y2, proj_w, proj_b, out);
}